// MessagePassing_2267742732507
// MI455X (gfx1250) — compile-verified
//
#include <hip/hip_runtime.h>

#define N_NODES 100000
#define N_EDGES 1600000
#define IN_F    128
#define OUT_F   128
#define KEEP_P  0.9f

typedef __attribute__((ext_vector_type(2))) float v2f;
typedef __attribute__((ext_vector_type(8))) float v8f;

// ---------------------------------------------------------------------------
// Kernel 0: one-time transpose of W (128x128, 64KB) so B fragments become
// contiguous float2 loads (global_load_b64 instead of 2x global_load_b32).
// ---------------------------------------------------------------------------
__global__ __launch_bounds__(256) void transpose_w(const float* __restrict__ W,
                                                   float* __restrict__ WT) {
  int i = blockIdx.x * 256 + threadIdx.x;   // 0 .. 16383
  int k = i >> 7;
  int n = i & 127;
  WT[n * IN_F + k] = W[i];                  // WT[n][k] = W[k][n]
}

// ---------------------------------------------------------------------------
// Kernel 1: H = X @ W + b  via V_WMMA_F32_16X16X4_F32 (fp32-exact matrix path)
// 256 threads = 8 wave32; wave w computes the 16x16 tile at cols [16w,16w+16)
// for the 16-row stripe at blockIdx.x*16. K loop: 32 x (16x16x4) WMMAs.
// Both A and B fragments are 8-byte vector loads.
// ---------------------------------------------------------------------------
__global__ __launch_bounds__(256) void gemm_bias_wmma(
    const float* __restrict__ X, const float* __restrict__ WT,
    const float* __restrict__ bias, float* __restrict__ H) {
  const int lane = threadIdx.x & 31;
  const int wave = threadIdx.x >> 5;
  const int m0   = blockIdx.x * 16;
  const int n0   = wave * 16;
  const int half = lane >> 4;   // 0: lanes 0-15, 1: lanes 16-31
  const int l16  = lane & 15;

  // A frag: a[v] = X[m0+l16][k0 + half*2 + v]        (contiguous pair)
  const float* xrow = X  + (size_t)(m0 + l16) * IN_F + half * 2;
  // B frag: b[v] = W[k0 + half*2 + v][n0+l16] = WT[n0+l16][k0 + half*2 + v]
  const float* wrow = WT + (size_t)(n0 + l16) * IN_F + half * 2;

  v8f c = {};
#pragma unroll
  for (int k0 = 0; k0 < IN_F; k0 += 4) {
    v2f a = *(const v2f*)(xrow + k0);
    v2f b = *(const v2f*)(wrow + k0);
    c = __builtin_amdgcn_wmma_f32_16x16x4_f32(
        /*neg_a=*/false, a, /*neg_b=*/false, b,
        /*c_mod=*/(short)0, c, /*reuse_a=*/false, /*reuse_b=*/false);
  }

  const float bv = bias[n0 + l16];
  // C/D layout: VGPR v -> row (v + half*8), col l16
#pragma unroll
  for (int v = 0; v < 8; ++v) {
    H[(size_t)(m0 + v + half * 8) * OUT_F + n0 + l16] = c[v] + bv;
  }
}

// ---------------------------------------------------------------------------
// Kernel 2: zero the accumulator (graph-replay-safe re-init of d_out)
// ---------------------------------------------------------------------------
__global__ __launch_bounds__(256) void zero_f4(float4* __restrict__ out, int n4) {
  int i = blockIdx.x * 256 + threadIdx.x;
  if (i < n4) out[i] = make_float4(0.f, 0.f, 0.f, 0.f);
}

// ---------------------------------------------------------------------------
// Kernel 3: edge gather->scale->scatter-add. One wave32 per edge; each lane
// moves a float4 slice of the 128-wide row (coalesced 512B per edge).
// H and the accumulator both fit in the 192MB L2, so the f32 atomics
// resolve in L2 rather than HBM.
// ---------------------------------------------------------------------------
__global__ __launch_bounds__(256) void edge_scatter(
    const float* __restrict__ H, const int* __restrict__ rows,
    const int* __restrict__ cols, const float* __restrict__ vals,
    float* __restrict__ out) {
  const int lane = threadIdx.x & 31;
  const int wave = threadIdx.x >> 5;
  const int e = blockIdx.x * 8 + wave;
  if (e >= N_EDGES) return;

  const int   r = rows[e];
  const int   s = cols[e];
  const float v = vals[e];

  const float4 h = *(const float4*)(H + (size_t)s * OUT_F + lane * 4);
  float* dst = out + (size_t)r * OUT_F + lane * 4;
  __hip_atomic_fetch_add(dst + 0, h.x * v, __ATOMIC_RELAXED, __HIP_MEMORY_SCOPE_AGENT);
  __hip_atomic_fetch_add(dst + 1, h.y * v, __ATOMIC_RELAXED, __HIP_MEMORY_SCOPE_AGENT);
  __hip_atomic_fetch_add(dst + 2, h.z * v, __ATOMIC_RELAXED, __HIP_MEMORY_SCOPE_AGENT);
  __hip_atomic_fetch_add(dst + 3, h.w * v, __ATOMIC_RELAXED, __HIP_MEMORY_SCOPE_AGENT);
}

// ---------------------------------------------------------------------------
// Kernel 4: ReLU + deterministic dropout (PCG hash of element index, p=0.1)
// ---------------------------------------------------------------------------
__device__ __forceinline__ unsigned pcg_hash(unsigned x) {
  x = x * 747796405u + 2891336453u;
  unsigned w = ((x >> ((x >> 28u) + 4u)) ^ x) * 277803737u;
  return (w >> 22u) ^ w;
}

__global__ __launch_bounds__(256) void relu_dropout(float* __restrict__ out, int n) {
  int i = (blockIdx.x * 256 + threadIdx.x) * 4;
  if (i >= n) return;
  const unsigned thresh = (unsigned)(KEEP_P * 4294967296.0);  // keep if hash < thresh
  const float inv_keep = 1.0f / KEEP_P;
  float4 x = *(float4*)(out + i);
  float r[4] = {x.x, x.y, x.z, x.w};
#pragma unroll
  for (int j = 0; j < 4; ++j) {
    float h = r[j] > 0.f ? r[j] : 0.f;                      // relu
    unsigned u = pcg_hash((unsigned)(i + j) ^ 0x2A2A2A2Au); // seed 42-ish
    r[j] = (u < thresh) ? h * inv_keep : 0.f;               // dropout
  }
  *(float4*)(out + i) = make_float4(r[0], r[1], r[2], r[3]);
}

// ---------------------------------------------------------------------------
extern "C" void kernel_launch(void* const* d_in, const int* in_sizes, int n_in,
                              void* d_out, int out_size, void* d_ws, size_t ws_size,
                              hipStream_t stream) {
  const float* X    = (const float*)d_in[0];  // [N_NODES, IN_F]
  const float* W    = (const float*)d_in[1];  // [IN_F, OUT_F]
  const float* bias = (const float*)d_in[2];  // [OUT_F]
  const int*   rows = (const int*)d_in[3];    // [N_EDGES]
  const int*   cols = (const int*)d_in[4];    // [N_EDGES]
  const float* vals = (const float*)d_in[5];  // [N_EDGES]
  float*       out  = (float*)d_out;          // [N_NODES, OUT_F]

  float* H  = (float*)d_ws;                   // [N_NODES, OUT_F]  (51.2 MB)
  float* WT = H + (size_t)N_NODES * OUT_F;    // [OUT_F, IN_F]     (64 KB)

  // 0) WT = W^T (once per launch; makes B frags contiguous b64 loads)
  transpose_w<<<(IN_F * OUT_F) / 256, 256, 0, stream>>>(W, WT);

  // 1) H = X @ W + b   (WMMA f32 path)
  gemm_bias_wmma<<<N_NODES / 16, 256, 0, stream>>>(X, WT, bias, H);

  // 2) out = 0
  const int n = N_NODES * OUT_F;              // 12.8M floats
  zero_f4<<<(n / 4 + 255) / 256, 256, 0, stream>>>((float4*)out, n / 4);

  // 3) out[rows] += H[cols] * vals
  edge_scatter<<<(N_EDGES + 7) / 8, 256, 0, stream>>>(H, rows, cols, vals, out);

  // 4) relu + dropout in place
  relu_dropout<<<(n / 4 + 255) / 256, 256, 0, stream>>>(out, n);
}